// LocalNorm_5368709120186
// MI455X (gfx1250) — compile-verified
//
#include <hip/hip_runtime.h>
#include <stdint.h>

// Local 33x33 normalization, fused single kernel.
// x: [N=512, 256, 256] f32 (N = B*C). One workgroup = one (image, row-band) tile.
//
// LDS layout (dynamic, 296,064 B <= 320 KB WGP LDS):
//   A: ROWS_A x STRIDE floats  -- holds x (with +-32-row halo), then centered c, then out
//   B: ROWS_B x STRIDE floats  -- holds vertical window sums (of x, then of c^2)
// STRIDE = 257 dwords -> bank = (row + col) % 64, conflict-free for both
// column-parallel and row-parallel stages. TDM pad feature creates this stride
// in hardware during the DMA load.

#define H       256
#define W       256
#define RAD     16            // window radius (33-tap)
#define CORE    96            // output rows per tile -> 3 tiles/image (96,96,64)
#define ROWS_A  (CORE + 4*RAD)   // 160
#define ROWS_B  (CORE + 2*RAD)   // 128
#define STRIDE  257
#define NTHREADS 256

typedef __attribute__((ext_vector_type(4))) unsigned int v4u;
typedef __attribute__((ext_vector_type(8))) int          v8i;
typedef __attribute__((ext_vector_type(4))) int          v4i;

#if defined(__AMDGCN__) && __has_builtin(__builtin_amdgcn_tensor_load_to_lds) && \
    __has_builtin(__builtin_amdgcn_tensor_store_from_lds) && \
    __has_builtin(__builtin_amdgcn_s_wait_tensorcnt)
#define USE_TDM 1
#else
#define USE_TDM 0
#endif

#if USE_TDM
// Pack D# group 1 per CDNA5 ISA (2-D tensor, dims 2+ unused).
__device__ __forceinline__ v8i tdm_group1(unsigned flags_w0, unsigned tensor_d0,
                                          unsigned tensor_d1, unsigned tile_d0,
                                          unsigned tile_d1, unsigned stride0) {
  v8i g;
  g[0] = (int)flags_w0;                                            // mask/data_size/pad
  g[1] = (int)((tensor_d0 & 0xFFFFu) << 16);                       // barrier addr 0 | d0.lo16
  g[2] = (int)((tensor_d0 >> 16) | ((tensor_d1 & 0xFFFFu) << 16)); // d0.hi16 | d1.lo16
  g[3] = (int)((tensor_d1 >> 16) | (tile_d0 << 16));               // d1.hi16 | tile_d0
  g[4] = (int)(tile_d1 & 0xFFFFu);                                 // tile_d1 | tile_d2=0
  g[5] = (int)stride0;                                             // dim0 stride lo32
  g[6] = 0;                                                        // stride0 hi | stride1 lo
  g[7] = 0;                                                        // stride1 hi (2-D: unused)
  return g;
}

__device__ __forceinline__ v4u tdm_group0(unsigned lds_addr, unsigned long long ga) {
  v4u g = { 1u,                                  // count=1 valid descriptor
            lds_addr,                            // LDS byte address
            (unsigned)ga,                        // global_addr[31:0]
            ((unsigned)(ga >> 32) & 0x1FFFFFFu)  // global_addr[56:32]
              | (2u << 30) };                    // type=2 ("image")
  return g;
}
#endif

__global__ void __launch_bounds__(NTHREADS)
localnorm_kernel(const float* __restrict__ x, float* __restrict__ out) {
  extern __shared__ float smem[];
  float* Abuf = smem;                          // ROWS_A * STRIDE
  float* Bbuf = smem + ROWS_A * STRIDE;        // ROWS_B * STRIDE

  const int tiles = (H + CORE - 1) / CORE;     // 3
  const int bid  = blockIdx.x;
  const int n    = bid / tiles;
  const int t0   = (bid - n * tiles) * CORE;   // first output row of tile
  const int core = min(CORE, H - t0);
  const int tid  = threadIdx.x;

  const float* img  = x   + (size_t)n * H * W;
  float*       outp = out + (size_t)n * H * W;

  const int gstart = t0 - 2 * RAD;             // global row of A row 0 (may be <0)
  const int zrows  = gstart < 0 ? -gstart : 0; // top rows to zero-fill
  const int lstart = gstart + zrows;           // first in-image row DMA'd
  const int arows  = core + 4 * RAD;           // A rows used this tile
  const int lrows  = arows - zrows;            // rows the DMA covers

  // Zero-fill top halo (first tile only); bottom halo zeros come from TDM OOB.
  for (int idx = tid; idx < zrows * W; idx += NTHREADS)
    Abuf[(idx >> 8) * STRIDE + (idx & (W - 1))] = 0.0f;

  // ---------------- Load x tile -> LDS A (hardware-padded to STRIDE) -------------
#if USE_TDM
  if (__builtin_amdgcn_readfirstlane(tid >> 5) == 0) {   // wave 0 only (scalar branch)
    unsigned long long ga = (unsigned long long)(const void*)(img + (size_t)lstart * W);
    unsigned lds = (unsigned)(unsigned long long)(const void*)&Abuf[zrows * STRIDE];
    // data_size=4B(2), pad_enable, pad_interval=7 (256 dw), pad_amount=0 (1 dw)
    unsigned w0 = (2u << 16) | (1u << 20) | (7u << 22);
    v4i gz4 = {0, 0, 0, 0};
    v8i gz8 = {0, 0, 0, 0, 0, 0, 0, 0};
    __builtin_amdgcn_tensor_load_to_lds(
        tdm_group0(lds, ga),
        tdm_group1(w0, W, (unsigned)(H - lstart) /* OOB rows -> zero */,
                   W, (unsigned)lrows, W),
        gz4, gz4, gz8, 0);
    __builtin_amdgcn_s_wait_tensorcnt(0);
  }
#else
  for (int idx = tid; idx < lrows * W; idx += NTHREADS) {
    int r = idx >> 8, w = idx & (W - 1);
    int g = lstart + r;
    Abuf[(zrows + r) * STRIDE + w] = (g < H) ? img[(size_t)g * W + w] : 0.0f;
  }
#endif
  __syncthreads();

  // ---- Stage 1: vertical 33-tap sliding sum of x, A -> B (core+32 rows) --------
  {
    const int w = tid;                         // one thread per column, conflict-free
    float s = 0.f;
    #pragma unroll 8
    for (int i = 0; i < 2 * RAD; ++i) s += Abuf[i * STRIDE + w];
    const int rows1 = core + 2 * RAD;
    for (int j = 0; j < rows1; ++j) {
      s += Abuf[(j + 2 * RAD) * STRIDE + w];
      Bbuf[j * STRIDE + w] = s;
      s -= Abuf[j * STRIDE + w];
    }
  }
  __syncthreads();

  // ---- Stage 1b: horizontal 33-tap of B -> mean; c = x - mean (overwrite A) ----
  {
    const int rows1 = core + 2 * RAD;
    const int j = tid & 127;                   // lanes -> distinct rows: conflict-free
    const int half = tid >> 7;                 // waves 0-3: cols 0-127, waves 4-7: 128-255
    if (j < rows1) {
      const int gh = t0 - RAD + j;             // global row of this B row
      const int w0 = half * (W / 2);
      float* Arow = &Abuf[(j + RAD) * STRIDE];
      if (gh < 0 || gh >= H) {
        // centered values outside the image are ZERO (reduce_window zero-pad)
        for (int i = 0; i < W / 2; ++i) Arow[w0 + i] = 0.0f;
      } else {
        const float cv = (float)(min(gh + RAD, H - 1) - max(gh - RAD, 0) + 1);
        const float* Brow = &Bbuf[j * STRIDE];
        float s = 0.f;
        for (int u = w0 - RAD; u < w0 + RAD; ++u)
          s += (u >= 0 && u < W) ? Brow[u] : 0.f;
        for (int i = 0; i < W / 2; ++i) {
          const int w = w0 + i;
          s += (w + RAD < W) ? Brow[w + RAD] : 0.f;
          const float ch = (float)(min(w + RAD, W - 1) - max(w - RAD, 0) + 1);
          const float mean = s / (cv * ch);
          Arow[w] = Arow[w] - mean;
          s -= (w - RAD >= 0) ? Brow[w - RAD] : 0.f;
        }
      }
    }
  }
  __syncthreads();

  // ---- Stage 2: vertical 33-tap sliding sum of c^2, A -> B (core rows) ---------
  {
    const int w = tid;
    float s = 0.f;
    #pragma unroll 8
    for (int i = RAD; i < 3 * RAD; ++i) { float v = Abuf[i * STRIDE + w]; s += v * v; }
    for (int j2 = 0; j2 < core; ++j2) {
      float v = Abuf[(j2 + 3 * RAD) * STRIDE + w]; s += v * v;
      Bbuf[j2 * STRIDE + w] = s;
      float u = Abuf[(j2 + RAD) * STRIDE + w];     s -= u * u;
    }
  }
  __syncthreads();

  // ---- Stage 2b: horizontal 33-tap of B -> S2; out = c/(sqrt(S2/cnt)+1e-8) -----
  {
    const int j2 = tid & 127;
    const int half = tid >> 7;
    if (j2 < core) {
      const int gh = t0 + j2;
      const float cv = (float)(min(gh + RAD, H - 1) - max(gh - RAD, 0) + 1);
      const int w0 = half * (W / 2);
      const float* Brow = &Bbuf[j2 * STRIDE];
      float* Arow = &Abuf[(j2 + 2 * RAD) * STRIDE];
      float s = 0.f;
      for (int u = w0 - RAD; u < w0 + RAD; ++u)
        s += (u >= 0 && u < W) ? Brow[u] : 0.f;
      for (int i = 0; i < W / 2; ++i) {
        const int w = w0 + i;
        s += (w + RAD < W) ? Brow[w + RAD] : 0.f;
        const float ch   = (float)(min(w + RAD, W - 1) - max(w - RAD, 0) + 1);
        const float stdv = sqrtf(s / (cv * ch));
        Arow[w] = Arow[w] / (stdv + 1e-8f);        // overwrite c with out
        s -= (w - RAD >= 0) ? Brow[w - RAD] : 0.f;
      }
    }
  }
  __syncthreads();

  // ---------------- Store out tile from LDS A (skip pad dword via OOB) ----------
#if USE_TDM
  if (__builtin_amdgcn_readfirstlane(tid >> 5) == 0) {
    unsigned long long ga = (unsigned long long)(void*)(outp + (size_t)t0 * W);
    unsigned lds = (unsigned)(unsigned long long)(void*)&Abuf[2 * RAD * STRIDE];
    unsigned w0 = (2u << 16);                      // data_size=4B, no pad on store
    v4i gz4 = {0, 0, 0, 0};
    v8i gz8 = {0, 0, 0, 0, 0, 0, 0, 0};
    // tile_dim0 = 257 > tensor_dim0 = 256: element X=256 (the LDS pad dword) is
    // OOB in dim0 and dropped, de-striding the padded LDS rows in hardware.
    __builtin_amdgcn_tensor_store_from_lds(
        tdm_group0(lds, ga),
        tdm_group1(w0, W, (unsigned)core, STRIDE, (unsigned)core, W),
        gz4, gz4, gz8, 0);
    __builtin_amdgcn_s_wait_tensorcnt(0);
  }
#else
  for (int idx = tid; idx < core * W; idx += NTHREADS) {
    int r = idx >> 8, w = idx & (W - 1);
    outp[(size_t)(t0 + r) * W + w] = Abuf[(r + 2 * RAD) * STRIDE + w];
  }
#endif
}

extern "C" void kernel_launch(void* const* d_in, const int* in_sizes, int n_in,
                              void* d_out, int out_size, void* d_ws, size_t ws_size,
                              hipStream_t stream) {
  (void)d_ws; (void)ws_size; (void)n_in; (void)out_size;
  const float* x = (const float*)d_in[0];
  float* out = (float*)d_out;
  const int n_imgs = in_sizes[0] / (H * W);                 // B*C = 512
  const int tiles = (H + CORE - 1) / CORE;                  // 3
  const size_t shmem = (size_t)(ROWS_A + ROWS_B) * STRIDE * sizeof(float); // 296,064 B
  // Opt in to >64KB dynamic LDS (gfx1250 WGP has 320KB).
  (void)hipFuncSetAttribute((const void*)localnorm_kernel,
                            hipFuncAttributeMaxDynamicSharedMemorySize, (int)shmem);
  hipLaunchKernelGGL(localnorm_kernel, dim3(n_imgs * tiles), dim3(NTHREADS),
                     shmem, stream, x, out);
}